// SSIM_6493990552040
// MI455X (gfx1250) — compile-verified
//
#include <hip/hip_runtime.h>

typedef __attribute__((ext_vector_type(2))) float v2f;
typedef __attribute__((ext_vector_type(8))) float v8f;

#define H_IMG 512
#define W_IMG 512
#define NPLANES 48                // 16 batch * 3 channels
#define TILE_H 16                 // output tile rows
#define TILE_W 64                 // output tile cols
#define TILES_X (W_IMG / TILE_W)  // 8
#define TILES_Y (H_IMG / TILE_H)  // 32
#define NBLOCKS (NPLANES * TILES_X * TILES_Y)
#define IN_ROWS 32                // 26 live rows padded to 2 M-groups of 16
#define IN_COLS 76                // 74 live cols padded for K=28 per col-group
#define H_ROWS 32
#define H_STRIDE 66               // 64 cols + 2 pad (LDS bank de-conflict)
#define NFIELDS 5

#if defined(__gfx1250__) &&                                         \
    __has_builtin(__builtin_amdgcn_global_load_async_to_lds_b32) && \
    __has_builtin(__builtin_amdgcn_s_wait_asynccnt)
#define SSIM_ASYNC 1
typedef __attribute__((address_space(1))) int GINT;  // global int
typedef __attribute__((address_space(3))) int LINT;  // LDS int
#else
#define SSIM_ASYNC 0
#endif

// 1D Gaussian, sigma=1.5, K=11, normalized (matches reference _gaussian_window)
__constant__ float c_gauss[11] = {
    0.00102846f, 0.00759876f, 0.03600077f, 0.10936079f, 0.21300552f,
    0.26601172f, 0.21300552f, 0.10936079f, 0.03600077f, 0.00759876f,
    0.00102846f};

__global__ __launch_bounds__(128) void ssim_tile_kernel(
    const float* __restrict__ img1, const float* __restrict__ img2,
    float* __restrict__ partial) {
  __shared__ float sA[IN_ROWS][IN_COLS];            // img1 tile (zero padded)
  __shared__ float sB[IN_ROWS][IN_COLS];            // img2 tile (zero padded)
  __shared__ float sH[NFIELDS][H_ROWS][H_STRIDE];   // horizontally convolved fields
  __shared__ float gfull[44];  // gfull[16+d]=g[d] for d in [0,10], else 0
  __shared__ float wsum[4];

  const int tid  = threadIdx.x;
  const int lane = tid & 31;
  const int wid  = __builtin_amdgcn_readfirstlane(tid >> 5);  // wave-uniform scalar
  const int m    = lane & 15;  // fragment row (A) / col (B) index
  const int half = lane >> 4;  // K-half selector per WMMA VGPR layout

  const int plane = blockIdx.y;
  const int tIdx  = blockIdx.x;
  const int tR = (tIdx / TILES_X) * TILE_H;
  const int tC = (tIdx % TILES_X) * TILE_W;
  const int R0 = tR - 5, C0 = tC - 5;

  const float* p1 = img1 + (size_t)plane * H_IMG * W_IMG;
  const float* p2 = img2 + (size_t)plane * H_IMG * W_IMG;

  for (int i = tid; i < 44; i += 128)
    gfull[i] = (i >= 16 && i <= 26) ? c_gauss[i - 16] : 0.0f;

  const bool interior =
      (R0 >= 0) && (C0 >= 0) && (R0 + 26 <= H_IMG) && (C0 + 74 <= W_IMG);

#if SSIM_ASYNC
  if (interior) {
    // zero padding strips: rows 26..31 (all cols) and cols 74..75 (rows 0..25)
    for (int i = tid; i < 6 * IN_COLS; i += 128) {
      int r = 26 + i / IN_COLS, c = i % IN_COLS;
      sA[r][c] = 0.0f;
      sB[r][c] = 0.0f;
    }
    for (int i = tid; i < 52; i += 128) {
      int r = i >> 1, c = 74 + (i & 1);
      sA[r][c] = 0.0f;
      sB[r][c] = 0.0f;
    }
    // async global->LDS stream of the 26x74 live region (no VGPR staging)
    for (int idx = tid; idx < 26 * 74; idx += 128) {
      int r = idx / 74, c = idx - r * 74;
      const float* g1 = p1 + (size_t)(R0 + r) * W_IMG + (C0 + c);
      const float* g2 = p2 + (size_t)(R0 + r) * W_IMG + (C0 + c);
      __builtin_amdgcn_global_load_async_to_lds_b32((GINT*)g1, (LINT*)&sA[r][c], 0, 0);
      __builtin_amdgcn_global_load_async_to_lds_b32((GINT*)g2, (LINT*)&sB[r][c], 0, 0);
    }
    __builtin_amdgcn_s_wait_asynccnt(0);
  } else
#endif
  {
    // guarded load path (border tiles / async-less fallback), SAME-zero padding
    for (int idx = tid; idx < IN_ROWS * IN_COLS; idx += 128) {
      int r = idx / IN_COLS, c = idx % IN_COLS;
      int gr = R0 + r, gc = C0 + c;
      bool ok = (r < 26) && (c < 74) && (gr >= 0) && (gr < H_IMG) &&
                (gc >= 0) && (gc < W_IMG);
      float a = 0.0f, b = 0.0f;
      if (ok) {
        size_t off = (size_t)gr * W_IMG + gc;
        a = p1[off];
        b = p2[off];
      }
      sA[r][c] = a;
      sB[r][c] = b;
    }
  }
  __syncthreads();

  // Per-lane banded-Gaussian fragments, shared by BOTH passes:
  // Wh[j][n]=g[j-n] with n=lane&15;  Wv[m][j]=g[j-m] with m=lane&15.
  v2f wband[7];
#pragma unroll
  for (int kc = 0; kc < 7; ++kc) {
    int j = 4 * kc + 2 * half;
    wband[kc] = {gfull[16 + j - m], gfull[16 + j + 1 - m]};
  }

  // ---- Horizontal pass: each wave owns col-group cg = wid; 2 row-groups;
  // all 5 fields fused so one (a,b) LDS load pair feeds 5 WMMAs.
#pragma unroll
  for (int rg = 0; rg < 2; ++rg) {
    const int row = 16 * rg + m;
    v8f h0 = {}, h1 = {}, h2 = {}, h3 = {}, h4 = {};
#pragma unroll
    for (int kc = 0; kc < 7; ++kc) {
      const int c0 = 16 * wid + 4 * kc + 2 * half;
      v2f a = {sA[row][c0], sA[row][c0 + 1]};
      v2f b = {sB[row][c0], sB[row][c0 + 1]};
      v2f aa = a * a, bb = b * b, ab = a * b;
      h0 = __builtin_amdgcn_wmma_f32_16x16x4_f32(false, a,  false, wband[kc], (short)0, h0, false, false);
      h1 = __builtin_amdgcn_wmma_f32_16x16x4_f32(false, b,  false, wband[kc], (short)0, h1, false, false);
      h2 = __builtin_amdgcn_wmma_f32_16x16x4_f32(false, aa, false, wband[kc], (short)0, h2, false, false);
      h3 = __builtin_amdgcn_wmma_f32_16x16x4_f32(false, bb, false, wband[kc], (short)0, h3, false, false);
      h4 = __builtin_amdgcn_wmma_f32_16x16x4_f32(false, ab, false, wband[kc], (short)0, h4, false, false);
    }
    const int hr = 16 * rg + 8 * half;  // D-layout: M = v + 8*half
    const int hc = 16 * wid + m;
#pragma unroll
    for (int v = 0; v < 8; ++v) {
      sH[0][hr + v][hc] = h0[v];
      sH[1][hr + v][hc] = h1[v];
      sH[2][hr + v][hc] = h2[v];
      sH[3][hr + v][hc] = h3[v];
      sH[4][hr + v][hc] = h4[v];
    }
  }
  __syncthreads();

  // ---- Vertical pass: Out(16x16) = Wv(16x28) x H(28x16); wave owns col-group
  // pc = wid; all 5 field accumulators resident -> SSIM computed in-register.
  v8f acc0 = {}, acc1 = {}, acc2 = {}, acc3 = {}, acc4 = {};
  const int cc = 16 * wid + m;
#pragma unroll
  for (int kc = 0; kc < 7; ++kc) {
    const int rr = 4 * kc + 2 * half;
    v2f B0 = {sH[0][rr][cc], sH[0][rr + 1][cc]};
    acc0 = __builtin_amdgcn_wmma_f32_16x16x4_f32(false, wband[kc], false, B0, (short)0, acc0, false, false);
    v2f B1 = {sH[1][rr][cc], sH[1][rr + 1][cc]};
    acc1 = __builtin_amdgcn_wmma_f32_16x16x4_f32(false, wband[kc], false, B1, (short)0, acc1, false, false);
    v2f B2 = {sH[2][rr][cc], sH[2][rr + 1][cc]};
    acc2 = __builtin_amdgcn_wmma_f32_16x16x4_f32(false, wband[kc], false, B2, (short)0, acc2, false, false);
    v2f B3 = {sH[3][rr][cc], sH[3][rr + 1][cc]};
    acc3 = __builtin_amdgcn_wmma_f32_16x16x4_f32(false, wband[kc], false, B3, (short)0, acc3, false, false);
    v2f B4 = {sH[4][rr][cc], sH[4][rr + 1][cc]};
    acc4 = __builtin_amdgcn_wmma_f32_16x16x4_f32(false, wband[kc], false, B4, (short)0, acc4, false, false);
  }

  // ---- Elementwise SSIM map + deterministic reduction
  const float C1c = 1e-4f, C2c = 9e-4f;
  float s = 0.0f;
#pragma unroll
  for (int v = 0; v < 8; ++v) {
    float mu1 = acc0[v], mu2 = acc1[v];
    float ex1 = acc2[v], ex2 = acc3[v], ex12 = acc4[v];
    float mu1s = mu1 * mu1, mu2s = mu2 * mu2, mu12 = mu1 * mu2;
    float s1 = ex1 - mu1s, s2 = ex2 - mu2s, s12 = ex12 - mu12;
    float v1  = 2.0f * s12 + C2c;
    float v2v = s1 + s2 + C2c + 1e-6f;
    s += (2.0f * mu12 + C1c) * v1 / ((mu1s + mu2s + C1c) * v2v);
  }
#pragma unroll
  for (int off = 16; off > 0; off >>= 1)
    s += __shfl_xor(s, off, 32);
  if (lane == 0) wsum[wid] = s;
  __syncthreads();
  if (tid == 0)
    partial[(size_t)blockIdx.y * gridDim.x + blockIdx.x] =
        wsum[0] + wsum[1] + wsum[2] + wsum[3];
}

__global__ __launch_bounds__(256) void ssim_reduce_kernel(
    const float* __restrict__ partial, int n, float* __restrict__ out,
    float scale) {
  __shared__ float wpart[8];
  float s = 0.0f;
  for (int i = threadIdx.x; i < n; i += 256) s += partial[i];
#pragma unroll
  for (int off = 16; off > 0; off >>= 1)
    s += __shfl_xor(s, off, 32);
  int lane = threadIdx.x & 31, wid = threadIdx.x >> 5;
  if (lane == 0) wpart[wid] = s;
  __syncthreads();
  if (threadIdx.x == 0) {
    float t = 0.0f;
#pragma unroll
    for (int i = 0; i < 8; ++i) t += wpart[i];
    out[0] = t * scale;
  }
}

extern "C" void kernel_launch(void* const* d_in, const int* in_sizes, int n_in,
                              void* d_out, int out_size, void* d_ws,
                              size_t ws_size, hipStream_t stream) {
  const float* img1 = (const float*)d_in[0];
  const float* img2 = (const float*)d_in[1];
  float* out = (float*)d_out;
  float* partial = (float*)d_ws;  // NBLOCKS floats = 48 KB scratch

  dim3 grid(TILES_X * TILES_Y, NPLANES);
  ssim_tile_kernel<<<grid, 128, 0, stream>>>(img1, img2, partial);

  const float scale = 1.0f / (float)((size_t)NPLANES * H_IMG * W_IMG);
  ssim_reduce_kernel<<<1, 256, 0, stream>>>(partial, NBLOCKS, out, scale);
}